// GATGraphRegressor_11553462026423
// MI455X (gfx1250) — compile-verified
//
#include <hip/hip_runtime.h>
#include <math.h>

typedef __attribute__((ext_vector_type(2))) float v2f;
typedef __attribute__((ext_vector_type(8))) float v8f;

#define NEG_SLOPE 0.2f

// ---------- helpers ----------
__device__ __forceinline__ float lrelu(float x) { return x > 0.f ? x : NEG_SLOPE * x; }
__device__ __forceinline__ float elu1(float x)  { return x > 0.f ? x : expm1f(x); }

// float -> order-preserving unsigned (for atomicMax on floats incl. negatives)
__device__ __forceinline__ unsigned f2ord(float f) {
  unsigned u = __float_as_uint(f);
  return (u & 0x80000000u) ? ~u : (u | 0x80000000u);
}
__device__ __forceinline__ float ord2f(unsigned u) {
  return (u & 0x80000000u) ? __uint_as_float(u & 0x7fffffffu) : __uint_as_float(~u);
}
#define ORD_NEG_INF 0x007FFFFFu  // f2ord(-inf)

__device__ __forceinline__ int edge_src(const int* ei, int e, int E) { return e < E ? ei[e] : e - E; }
__device__ __forceinline__ int edge_dst(const int* ei, int e, int E) { return e < E ? ei[E + e] : e - E; }

// ---------- init workspace ----------
__global__ void init_ws_k(float* acc1, float* den1, unsigned* mord1,
                          float* acc2, float* den2, unsigned* mord2,
                          float* psum, float* pcnt, int n_nodes) {
  long long t = (long long)blockIdx.x * blockDim.x + threadIdx.x;
  long long big = (long long)n_nodes * 256;
  if (t < big) acc1[t] = 0.f;
  if (t < (long long)n_nodes * 64) acc2[t] = 0.f;
  if (t < (long long)n_nodes * 4) { den1[t] = 0.f; mord1[t] = ORD_NEG_INF; }
  if (t < n_nodes) { den2[t] = 0.f; mord2[t] = ORD_NEG_INF; }
  if (t < 64 * 64) psum[t] = 0.f;
  if (t < 64) pcnt[t] = 0.f;
}

// ---------- fp32 WMMA GEMM: C[M,Ncol] = A[M,K] @ B[K,Ncol] ----------
// One wave32 computes one 16x16 tile using V_WMMA_F32_16X16X4_F32.
// A lane l holds A rows m = l&15, K pair starting at 2*(l>>4); B/D are
// row-striped across lanes (B: VGPR v -> k row v + 2*(l>>4); D: VGPR v -> row v + 8*(l>>4)).
// Out-of-range A rows are address-clamped (not zeroed): A row m only feeds
// D row m, and the store loop masks rows >= M, so garbage never escapes.
__global__ void wmma_gemm_f32_k(const float* __restrict__ A, const float* __restrict__ B,
                                float* __restrict__ C, int M, int Ncol, int K) {
  int tid  = blockIdx.x * blockDim.x + threadIdx.x;
  int wave = tid >> 5;            // wave-uniform
  int lane = threadIdx.x & 31;
  int mt = (M + 15) >> 4;
  int nt = Ncol >> 4;             // Ncol is a multiple of 16 here (256 or 64)
  int tm = wave / nt;
  int tn = wave % nt;
  if (tm >= mt) return;           // uniform per wave -> EXEC all-ones below

  int khalf = lane >> 4;          // 0: lanes 0-15, 1: lanes 16-31
  int mrow  = (tm << 4) + (lane & 15);
  int ncol  = (tn << 4) + (lane & 15);
  int mclmp = mrow < M ? mrow : 0;

  const float* ap = A + (size_t)mclmp * K + 2 * khalf;        // += 4 per step
  const float* bp = B + (size_t)(2 * khalf) * Ncol + ncol;    // += 4*Ncol per step
  const size_t bstep = (size_t)4 * Ncol;

  v8f acc = {};
#pragma unroll 4
  for (int k = 0; k < K; k += 4) {
    v2f a = *(const v2f*)ap;      // contiguous K pair -> global_load_b64
    v2f b;
    b.x = bp[0];
    b.y = bp[Ncol];
    acc = __builtin_amdgcn_wmma_f32_16x16x4_f32(false, a, false, b, (short)0, acc,
                                                false, false);
    ap += 4;
    bp += bstep;
  }
  // D layout: VGPR v -> row (tm*16 + v + 8*khalf), col = ncol
  int base = (tm << 4) + 8 * khalf;
#pragma unroll
  for (int v = 0; v < 8; ++v) {
    int row = base + v;
    if (row < M) C[(size_t)row * Ncol + ncol] = acc[v];
  }
}

// ---------- per-(node,head) attention score dots ----------
__global__ void scores_k(const float* __restrict__ hb, const float* __restrict__ asrc,
                         const float* __restrict__ adst, float* __restrict__ s,
                         float* __restrict__ d, int n_nodes, int H) {
  int t = blockIdx.x * blockDim.x + threadIdx.x;
  if (t >= n_nodes * H) return;
  int n = t / H, h = t % H;
  const float* row = hb + (size_t)n * H * 64 + (size_t)h * 64;
  const float* as = asrc + h * 64;
  const float* ad = adst + h * 64;
  float ss = 0.f, dd = 0.f;
#pragma unroll 8
  for (int c = 0; c < 64; ++c) { ss += row[c] * as[c]; dd += row[c] * ad[c]; }
  s[t] = ss; d[t] = dd;
}

// ---------- edge pass 1: segment max ----------
__global__ void edge_max_k(const int* __restrict__ ei, int E, int Etot,
                           const float* __restrict__ s, const float* __restrict__ d,
                           unsigned* __restrict__ mord, int H) {
  int e = blockIdx.x * blockDim.x + threadIdx.x;
  if (e >= Etot) return;
  int u = edge_src(ei, e, E), v = edge_dst(ei, e, E);
  for (int h = 0; h < H; ++h) {
    float l = lrelu(s[u * H + h] + d[v * H + h]);
    atomicMax(&mord[v * H + h], f2ord(l));
  }
}

// ---------- edge pass 2: denom = segment_sum(exp(logit - m)) ----------
__global__ void edge_sum_k(const int* __restrict__ ei, int E, int Etot,
                           const float* __restrict__ s, const float* __restrict__ d,
                           const unsigned* __restrict__ mord, float* __restrict__ den,
                           int H) {
  int e = blockIdx.x * blockDim.x + threadIdx.x;
  if (e >= Etot) return;
  int u = edge_src(ei, e, E), v = edge_dst(ei, e, E);
  for (int h = 0; h < H; ++h) {
    float m = ord2f(mord[v * H + h]);
    if (!isfinite(m)) m = 0.f;
    float l = lrelu(s[u * H + h] + d[v * H + h]);
    atomicAdd(&den[v * H + h], expf(l - m));
  }
}

// ---------- edge pass 3: acc[dst] += alpha * h[src]  (thread per edge x channel) ----------
__global__ void edge_agg_k(const int* __restrict__ ei, int E, int Etot,
                           const float* __restrict__ s, const float* __restrict__ d,
                           const unsigned* __restrict__ mord, const float* __restrict__ den,
                           const float* __restrict__ hb, float* __restrict__ acc, int H) {
  long long t = (long long)blockIdx.x * blockDim.x + threadIdx.x;
  long long tot = (long long)Etot * 64;
  if (t >= tot) return;
  int c = (int)(t & 63);
  int e = (int)(t >> 6);
  int u = edge_src(ei, e, E), v = edge_dst(ei, e, E);
  for (int h = 0; h < H; ++h) {
    float m = ord2f(mord[v * H + h]);
    if (!isfinite(m)) m = 0.f;
    float l = lrelu(s[u * H + h] + d[v * H + h]);
    float alpha = expf(l - m) / den[v * H + h];
    atomicAdd(&acc[(size_t)v * H * 64 + (size_t)h * 64 + c],
              alpha * hb[(size_t)u * H * 64 + (size_t)h * 64 + c]);
  }
}

// ---------- bias + ELU (in place) ----------
__global__ void bias_elu_k(float* __restrict__ acc, const float* __restrict__ b,
                           int n_nodes, int HC) {
  long long t = (long long)blockIdx.x * blockDim.x + threadIdx.x;
  if (t >= (long long)n_nodes * HC) return;
  int j = (int)(t % HC);
  acc[t] = elu1(acc[t] + b[j]);
}

// ---------- global mean pool accumulation ----------
__global__ void pool_k(const float* __restrict__ feat, const int* __restrict__ batch,
                       float* __restrict__ psum, float* __restrict__ pcnt, int n_nodes) {
  long long t = (long long)blockIdx.x * blockDim.x + threadIdx.x;
  if (t >= (long long)n_nodes * 64) return;
  int c = (int)(t & 63);
  int n = (int)(t >> 6);
  int g = batch[n];
  atomicAdd(&psum[g * 64 + c], feat[(size_t)n * 64 + c]);
  if (c == 0) atomicAdd(&pcnt[g], 1.0f);
}

// ---------- final linear: out[g] = (sum_c psum[g,c]*Wlin[c]) / max(cnt,1) + b ----------
__global__ void final_k(const float* __restrict__ psum, const float* __restrict__ pcnt,
                        const float* __restrict__ Wlin, const float* __restrict__ blin,
                        float* __restrict__ out) {
  int g = threadIdx.x;
  if (g >= 64) return;
  float acc = 0.f;
#pragma unroll 8
  for (int c = 0; c < 64; ++c) acc += psum[g * 64 + c] * Wlin[c];
  float cnt = pcnt[g];
  out[g] = acc / fmaxf(cnt, 1.0f) + blin[0];
}

extern "C" void kernel_launch(void* const* d_in, const int* in_sizes, int n_in,
                              void* d_out, int out_size, void* d_ws, size_t ws_size,
                              hipStream_t stream) {
  const float* x      = (const float*)d_in[0];
  const int*   ei     = (const int*)d_in[1];
  const int*   batch  = (const int*)d_in[2];
  const float* W1     = (const float*)d_in[3];
  const float* a_src1 = (const float*)d_in[4];
  const float* a_dst1 = (const float*)d_in[5];
  const float* b1     = (const float*)d_in[6];
  const float* W2     = (const float*)d_in[7];
  const float* a_src2 = (const float*)d_in[8];
  const float* a_dst2 = (const float*)d_in[9];
  const float* b2     = (const float*)d_in[10];
  const float* W_lin  = (const float*)d_in[11];
  const float* b_lin  = (const float*)d_in[12];
  float* out = (float*)d_out;

  const int Nn   = in_sizes[0] / 128;   // 20000 nodes
  const int Ee   = in_sizes[1] / 2;     // 320000 edges
  const int Etot = Ee + Nn;             // + self loops
  const int H1h  = 4;

  // ---- workspace layout (floats) ----
  float* w = (float*)d_ws;
  float* h1buf = w;                 w += (size_t)Nn * 256;
  float* acc1  = w;                 w += (size_t)Nn * 256;
  float* s1    = w;                 w += (size_t)Nn * 4;
  float* d1    = w;                 w += (size_t)Nn * 4;
  float* den1  = w;                 w += (size_t)Nn * 4;
  unsigned* mord1 = (unsigned*)w;   w += (size_t)Nn * 4;
  float* h2buf = w;                 w += (size_t)Nn * 64;
  float* acc2  = w;                 w += (size_t)Nn * 64;
  float* s2    = w;                 w += (size_t)Nn;
  float* d2    = w;                 w += (size_t)Nn;
  float* den2  = w;                 w += (size_t)Nn;
  unsigned* mord2 = (unsigned*)w;   w += (size_t)Nn;
  float* psum  = w;                 w += 64 * 64;
  float* pcnt  = w;                 w += 64;

  const int B = 256;
  auto blocks = [](long long total, int b) { return (unsigned)((total + b - 1) / b); };

  // 1) init accumulators / maxima
  init_ws_k<<<blocks((long long)Nn * 256, B), B, 0, stream>>>(
      acc1, den1, mord1, acc2, den2, mord2, psum, pcnt, Nn);

  // 2) GEMM1: h1 = x @ W1  (M=Nn, N=256, K=128)
  {
    long long waves = (long long)((Nn + 15) / 16) * (256 / 16);
    wmma_gemm_f32_k<<<blocks(waves * 32, B), B, 0, stream>>>(x, W1, h1buf, Nn, 256, 128);
  }
  // 3) scores
  scores_k<<<blocks((long long)Nn * H1h, B), B, 0, stream>>>(h1buf, a_src1, a_dst1, s1, d1, Nn, H1h);
  // 4-6) attention conv1
  edge_max_k<<<blocks(Etot, B), B, 0, stream>>>(ei, Ee, Etot, s1, d1, mord1, H1h);
  edge_sum_k<<<blocks(Etot, B), B, 0, stream>>>(ei, Ee, Etot, s1, d1, mord1, den1, H1h);
  edge_agg_k<<<blocks((long long)Etot * 64, B), B, 0, stream>>>(
      ei, Ee, Etot, s1, d1, mord1, den1, h1buf, acc1, H1h);
  // 7) bias + ELU
  bias_elu_k<<<blocks((long long)Nn * 256, B), B, 0, stream>>>(acc1, b1, Nn, 256);

  // 8) GEMM2: h2 = acc1 @ W2  (M=Nn, N=64, K=256)
  {
    long long waves = (long long)((Nn + 15) / 16) * (64 / 16);
    wmma_gemm_f32_k<<<blocks(waves * 32, B), B, 0, stream>>>(acc1, W2, h2buf, Nn, 64, 256);
  }
  // 9) scores (H=1)
  scores_k<<<blocks(Nn, B), B, 0, stream>>>(h2buf, a_src2, a_dst2, s2, d2, Nn, 1);
  // 10-12) attention conv2
  edge_max_k<<<blocks(Etot, B), B, 0, stream>>>(ei, Ee, Etot, s2, d2, mord2, 1);
  edge_sum_k<<<blocks(Etot, B), B, 0, stream>>>(ei, Ee, Etot, s2, d2, mord2, den2, 1);
  edge_agg_k<<<blocks((long long)Etot * 64, B), B, 0, stream>>>(
      ei, Ee, Etot, s2, d2, mord2, den2, h2buf, acc2, 1);
  // 13) bias + ELU
  bias_elu_k<<<blocks((long long)Nn * 64, B), B, 0, stream>>>(acc2, b2, Nn, 64);

  // 14) mean pool accumulate
  pool_k<<<blocks((long long)Nn * 64, B), B, 0, stream>>>(acc2, batch, psum, pcnt, Nn);
  // 15) final linear
  final_k<<<1, 64, 0, stream>>>(psum, pcnt, W_lin, b_lin, out);
}